// GraphLayer_61950608277594
// MI455X (gfx1250) — compile-verified
//
#include <hip/hip_runtime.h>
#include <math.h>

// ---------------------------------------------------------------------------
// GraphLayer on MI455X (gfx1250): f32 WMMA (16x16x4) for all GEMMs,
// VALU loop for the 2->256->1 edge MLP, fused softmax/LN via LDS.
// B=4, N=512, D=H=256
// ---------------------------------------------------------------------------

typedef __attribute__((ext_vector_type(2))) float v2f;
typedef __attribute__((ext_vector_type(8))) float v8f;

#define NB 4
#define NN 512
#define HD 256
#define LDA 258    // padded LDS stride for 256-wide tiles (bank-conflict free)
#define LDSS 514   // padded LDS stride for 512-wide tile

__device__ __forceinline__ v8f wmma_f32(v2f a, v2f b, v8f c) {
  // D = A(16x4 f32) x B(4x16 f32) + C(16x16 f32)
  return __builtin_amdgcn_wmma_f32_16x16x4_f32(
      /*neg_a=*/false, a, /*neg_b=*/false, b,
      /*c_mod=*/(short)0, c, /*reuse_a=*/false, /*reuse_b=*/false);
}

// ---------------------------------------------------------------------------
// Kernel 1: q/k/v projections.  One block = (batch, 16-row tile, which matrix).
// ---------------------------------------------------------------------------
__global__ void __launch_bounds__(256) qkv_kernel(
    const float* __restrict__ x,
    const float* __restrict__ Wq, const float* __restrict__ bq,
    const float* __restrict__ Wk, const float* __restrict__ bk,
    const float* __restrict__ Wv, const float* __restrict__ bv,
    float* __restrict__ q, float* __restrict__ k, float* __restrict__ v)
{
  __shared__ float xt[16 * LDA];
  int bid = blockIdx.x;
  int m  = bid % 3;
  int t  = bid / 3;
  int nt = t % (NN / 16);
  int b  = t / (NN / 16);
  int n0 = nt * 16;

  const float* W; const float* bias; float* out;
  if (m == 0)      { W = Wq; bias = bq; out = q; }
  else if (m == 1) { W = Wk; bias = bk; out = k; }
  else             { W = Wv; bias = bv; out = v; }

  int tid = threadIdx.x;
  // stage 16x256 x-tile into padded LDS
  for (int idx = tid; idx < 16 * HD; idx += 256) {
    int r = idx >> 8, c = idx & 255;
    xt[r * LDA + c] = x[((b * NN + n0 + r) * HD) + c];
  }
  __syncthreads();

  int wave = tid >> 5, lane = tid & 31;
  int ml = lane & 15, hl = lane >> 4;

  for (int hh = 0; hh < 2; ++hh) {
    int h0 = (wave * 2 + hh) * 16;
    v8f acc = {};
    const float* arow = xt + ml * LDA + 2 * hl;
    const float* wrow = W + (h0 + ml) * HD + 2 * hl;
#pragma unroll 4
    for (int k0 = 0; k0 < HD; k0 += 4) {
      v2f a  = *(const v2f*)(arow + k0);
      v2f bb = *(const v2f*)(wrow + k0);
      acc = wmma_f32(a, bb, acc);
    }
    float bsc = bias[h0 + ml];
#pragma unroll
    for (int r = 0; r < 8; ++r) {
      int row = r + 8 * hl;
      out[((b * NN + n0 + row) * HD) + h0 + ml] = acc[r] + bsc;
    }
  }
}

// ---------------------------------------------------------------------------
// Kernel 2: edge-weight MLP: ew[b,i,j] = sigmoid(We2 . relu(We1 @ [a,c] + be1) + be2)
// One block per (b,i); weights broadcast from LDS; each thread does 2 j's.
// ---------------------------------------------------------------------------
__global__ void __launch_bounds__(256) edge_kernel(
    const float* __restrict__ area, const float* __restrict__ co,
    const float* __restrict__ We1, const float* __restrict__ be1,
    const float* __restrict__ We2, const float* __restrict__ be2,
    float* __restrict__ ew)
{
  __shared__ float w01[2 * HD];
  __shared__ float wb[HD];
  __shared__ float w2[HD];
  int tid = threadIdx.x;
  for (int idx = tid; idx < 4 * HD; idx += 256) {
    if (idx < 2 * HD)       w01[idx] = We1[idx];
    else if (idx < 3 * HD)  wb[idx - 2 * HD] = be1[idx - 2 * HD];
    else                    w2[idx - 3 * HD] = We2[idx - 3 * HD];
  }
  __syncthreads();

  int b = blockIdx.x / NN, i = blockIdx.x % NN;
  float a  = area[b * NN + i];
  float c0 = co[i * NN + tid];
  float c1 = co[i * NN + tid + 256];
  float acc0 = 0.f, acc1 = 0.f;
  for (int h = 0; h < HD; ++h) {
    float w0 = w01[2 * h], w1 = w01[2 * h + 1];
    float base = fmaf(a, w0, wb[h]);
    float t0 = fmaxf(fmaf(c0, w1, base), 0.f);
    float t1 = fmaxf(fmaf(c1, w1, base), 0.f);
    float ww = w2[h];
    acc0 = fmaf(t0, ww, acc0);
    acc1 = fmaf(t1, ww, acc1);
  }
  float bb = be2[0];
  float e0 = 1.f / (1.f + __expf(-(acc0 + bb)));
  float e1 = 1.f / (1.f + __expf(-(acc1 + bb)));
  ew[((b * NN + i) * NN) + tid]       = e0;
  ew[((b * NN + i) * NN) + tid + 256] = e1;
}

// ---------------------------------------------------------------------------
// Kernel 3: scores -> *ew -> softmax -> @v -> LN1 -> @Wo -> LN+relu -> +x -> LN2
// One block per (b, 16 i-rows). All GEMMs via WMMA; reductions via shfl_xor.
// ---------------------------------------------------------------------------
__global__ void __launch_bounds__(256) attn_kernel(
    const float* __restrict__ q, const float* __restrict__ kmat,
    const float* __restrict__ vmat, const float* __restrict__ ew,
    const float* __restrict__ x,
    const float* __restrict__ Wo, const float* __restrict__ bo,
    const float* __restrict__ g1, const float* __restrict__ b1,
    const float* __restrict__ g_ot, const float* __restrict__ b_ot,
    const float* __restrict__ g2, const float* __restrict__ b2,
    float* __restrict__ out)
{
  __shared__ float at[16 * LDA];    // q tile, later attn-out / ln1
  __shared__ float st[16 * LDSS];   // scores/probs, later out2 (stride LDA)

  int b = blockIdx.x >> 5, it = blockIdx.x & 31;
  int i0 = it * 16;
  int tid = threadIdx.x, wave = tid >> 5, lane = tid & 31;
  int ml = lane & 15, hl = lane >> 4;

  // stage q tile
  for (int idx = tid; idx < 16 * HD; idx += 256) {
    int r = idx >> 8, c = idx & 255;
    at[r * LDA + c] = q[((b * NN + i0 + r) * HD) + c];
  }
  __syncthreads();

  // --- scores = q.k^T / sqrt(H) : each wave 4 j-tiles of 16 ---
  for (int jj = 0; jj < 4; ++jj) {
    int j0 = (wave * 4 + jj) * 16;
    v8f acc = {};
    const float* arow = at + ml * LDA + 2 * hl;
    const float* krow = kmat + ((b * NN + j0 + ml) * HD) + 2 * hl;
#pragma unroll 4
    for (int k0 = 0; k0 < HD; k0 += 4) {
      v2f a  = *(const v2f*)(arow + k0);
      v2f bb = *(const v2f*)(krow + k0);
      acc = wmma_f32(a, bb, acc);
    }
#pragma unroll
    for (int r = 0; r < 8; ++r)
      st[(r + 8 * hl) * LDSS + j0 + ml] = acc[r] * 0.0625f;  // 1/sqrt(256)
  }
  __syncthreads();

  // --- multiply by edge weights (coalesced) ---
  for (int idx = tid; idx < 16 * NN; idx += 256) {
    int r = idx >> 9, j = idx & (NN - 1);
    st[r * LDSS + j] *= ew[((b * NN + i0 + r) * NN) + j];
  }
  __syncthreads();

  // --- softmax over j (each wave owns 2 rows) ---
  for (int rr = 0; rr < 2; ++rr) {
    int row = wave * 2 + rr;
    float* srow = st + row * LDSS;
    float m = -3.4e38f;
    for (int t = lane; t < NN; t += 32) m = fmaxf(m, srow[t]);
    for (int off = 16; off > 0; off >>= 1) m = fmaxf(m, __shfl_xor(m, off, 32));
    float s = 0.f;
    for (int t = lane; t < NN; t += 32) {
      float e = __expf(srow[t] - m); srow[t] = e; s += e;
    }
    for (int off = 16; off > 0; off >>= 1) s += __shfl_xor(s, off, 32);
    float inv = 1.f / s;
    for (int t = lane; t < NN; t += 32) srow[t] *= inv;
  }
  __syncthreads();

  // --- attn_probs @ v : K over j=512, each wave 2 h-tiles ---
  for (int hh = 0; hh < 2; ++hh) {
    int h0 = (wave * 2 + hh) * 16;
    v8f acc = {};
    const float* arow = st + ml * LDSS + 2 * hl;
    const float* vcol = vmat + (size_t)b * NN * HD + h0 + ml;
#pragma unroll 2
    for (int k0 = 0; k0 < NN; k0 += 4) {
      v2f a = *(const v2f*)(arow + k0);
      int kj = k0 + 2 * hl;
      v2f bb; bb.x = vcol[kj * HD]; bb.y = vcol[(kj + 1) * HD];
      acc = wmma_f32(a, bb, acc);
    }
#pragma unroll
    for (int r = 0; r < 8; ++r)
      at[(r + 8 * hl) * LDA + h0 + ml] = acc[r];
  }
  __syncthreads();

  // --- LayerNorm(g1,b1) in place on `at` rows ---
  for (int rr = 0; rr < 2; ++rr) {
    int row = wave * 2 + rr;
    float* arow = at + row * LDA;
    float s = 0.f;
    for (int t = lane; t < HD; t += 32) s += arow[t];
    for (int off = 16; off > 0; off >>= 1) s += __shfl_xor(s, off, 32);
    float mean = s * (1.f / HD);
    float vs = 0.f;
    for (int t = lane; t < HD; t += 32) { float d = arow[t] - mean; vs += d * d; }
    for (int off = 16; off > 0; off >>= 1) vs += __shfl_xor(vs, off, 32);
    float rstd = rsqrtf(vs * (1.f / HD) + 1e-5f);
    for (int t = lane; t < HD; t += 32)
      arow[t] = (arow[t] - mean) * rstd * g1[t] + b1[t];
  }
  __syncthreads();

  // --- ln1 @ Wo^T + bo -> st (stride LDA) ---
  for (int oo = 0; oo < 2; ++oo) {
    int o0 = (wave * 2 + oo) * 16;
    v8f acc = {};
    const float* arow = at + ml * LDA + 2 * hl;
    const float* wrow = Wo + (o0 + ml) * HD + 2 * hl;
#pragma unroll 4
    for (int k0 = 0; k0 < HD; k0 += 4) {
      v2f a  = *(const v2f*)(arow + k0);
      v2f bb = *(const v2f*)(wrow + k0);
      acc = wmma_f32(a, bb, acc);
    }
    float bsc = bo[o0 + ml];
#pragma unroll
    for (int r = 0; r < 8; ++r)
      st[(r + 8 * hl) * LDA + o0 + ml] = acc[r] + bsc;
  }
  __syncthreads();

  // --- LN(g_ot,b_ot) + relu + residual(+x) + LN(g2,b2) -> out ---
  for (int rr = 0; rr < 2; ++rr) {
    int row = wave * 2 + rr;
    float* srow = st + row * LDA;
    const float* xrow = x + ((b * NN + i0 + row) * HD);
    float s = 0.f;
    for (int t = lane; t < HD; t += 32) s += srow[t];
    for (int off = 16; off > 0; off >>= 1) s += __shfl_xor(s, off, 32);
    float mean = s * (1.f / HD);
    float vs = 0.f;
    for (int t = lane; t < HD; t += 32) { float d = srow[t] - mean; vs += d * d; }
    for (int off = 16; off > 0; off >>= 1) vs += __shfl_xor(vs, off, 32);
    float rstd = rsqrtf(vs * (1.f / HD) + 1e-5f);
    float zs = 0.f;
    for (int t = lane; t < HD; t += 32) {
      float y = (srow[t] - mean) * rstd * g_ot[t] + b_ot[t];
      y = fmaxf(y, 0.f);
      float z = y + xrow[t];
      srow[t] = z;
      zs += z;
    }
    for (int off = 16; off > 0; off >>= 1) zs += __shfl_xor(zs, off, 32);
    float mean2 = zs * (1.f / HD);
    float vz = 0.f;
    for (int t = lane; t < HD; t += 32) { float d = srow[t] - mean2; vz += d * d; }
    for (int off = 16; off > 0; off >>= 1) vz += __shfl_xor(vz, off, 32);
    float rstd2 = rsqrtf(vz * (1.f / HD) + 1e-5f);
    float* orow = out + ((b * NN + i0 + row) * HD);
    for (int t = lane; t < HD; t += 32)
      orow[t] = (srow[t] - mean2) * rstd2 * g2[t] + b2[t];
  }
}

// ---------------------------------------------------------------------------
extern "C" void kernel_launch(void* const* d_in, const int* in_sizes, int n_in,
                              void* d_out, int out_size, void* d_ws, size_t ws_size,
                              hipStream_t stream) {
  const float* x    = (const float*)d_in[0];
  const float* area = (const float*)d_in[1];
  const float* co   = (const float*)d_in[2];
  const float* Wq   = (const float*)d_in[3];
  const float* bq   = (const float*)d_in[4];
  const float* Wk   = (const float*)d_in[5];
  const float* bk   = (const float*)d_in[6];
  const float* Wv   = (const float*)d_in[7];
  const float* bv   = (const float*)d_in[8];
  const float* We1  = (const float*)d_in[9];
  const float* be1  = (const float*)d_in[10];
  const float* We2  = (const float*)d_in[11];
  const float* be2  = (const float*)d_in[12];
  const float* Wo   = (const float*)d_in[13];
  const float* bo   = (const float*)d_in[14];
  const float* g_ot = (const float*)d_in[15];
  const float* b_ot = (const float*)d_in[16];
  const float* g1   = (const float*)d_in[17];
  const float* b1   = (const float*)d_in[18];
  const float* g2   = (const float*)d_in[19];
  const float* b2   = (const float*)d_in[20];
  float* outp = (float*)d_out;

  size_t need = ((size_t)3 * NB * NN * HD + (size_t)NB * NN * NN) * sizeof(float);
  if (ws_size < need) return;  // deterministic no-op if workspace too small
  float* q  = (float*)d_ws;
  float* k  = q + (size_t)NB * NN * HD;
  float* v  = k + (size_t)NB * NN * HD;
  float* ew = v + (size_t)NB * NN * HD;

  qkv_kernel<<<NB * (NN / 16) * 3, 256, 0, stream>>>(x, Wq, bq, Wk, bk, Wv, bv, q, k, v);
  edge_kernel<<<NB * NN, 256, 0, stream>>>(area, co, We1, be1, We2, be2, ew);
  attn_kernel<<<NB * (NN / 16), 256, 0, stream>>>(q, k, v, ew, x, Wo, bo,
                                                  g1, b1, g_ot, b_ot, g2, b2, outp);
}